// GPT5MoERouter_41824391528973
// MI455X (gfx1250) — compile-verified
//
#include <hip/hip_runtime.h>
#include <math.h>

typedef __attribute__((ext_vector_type(2))) float v2f;
typedef __attribute__((ext_vector_type(8))) float v8f;

#define H 1024
#define E 64
#define PANEL_K 128
#define WROW 132            // 128 + 4 pad floats -> conflict-free LDS
#define LROW 68             // 64 + 4 pad floats (slot 64 = 1/sum)
#define WPANEL_FLOATS (E * WROW)          // 8448
#define LOGITS_FLOATS (4 * 16 * LROW)     // 4352
#define SMEM_BYTES ((WPANEL_FLOATS + LOGITS_FLOATS) * 4)   // 51200 B

// ---------------- Kernel 1: WMMA router GEMM + softmax + top2 ----------------
// block = 128 threads (4 waves), each wave owns a 16-token tile, all 64 experts.
__global__ __launch_bounds__(128)
void router_fused_kernel(const float* __restrict__ x,
                         const float* __restrict__ W,
                         const float* __restrict__ b,
                         float* __restrict__ ew,        // [T,2]
                         float* __restrict__ ei,        // [T,2] (as float)
                         float* __restrict__ probs,     // [T,64]
                         unsigned* __restrict__ topPacked) {
    extern __shared__ float smem[];
    float* wp   = smem;                     // W panel: 64 x WROW
    float* slog = smem + WPANEL_FLOATS;     // 4 waves x 16 x LROW

    const int tid  = threadIdx.x;
    const int wave = tid >> 5;
    const int lane = tid & 31;
    const int r    = lane & 15;
    const int half = lane >> 4;
    const int koff = half * 2;
    const int tokBase = blockIdx.x * 64 + wave * 16;

    v8f acc[4] = {};

    const float* xrow = x + (size_t)(tokBase + r) * H + koff;

    for (int kp = 0; kp < H; kp += PANEL_K) {
        __syncthreads();
        // Cooperative stage of W panel [64 x 128] into padded LDS.
        for (int i = tid; i < E * (PANEL_K / 4); i += 128) {
            int row = i >> 5, c4 = i & 31;
            float4 v = *(const float4*)(W + row * H + kp + c4 * 4);
            *(float4*)(wp + row * WROW + c4 * 4) = v;
        }
        __syncthreads();

        if (kp + PANEL_K < H)
            __builtin_prefetch(xrow + kp + PANEL_K, 0, 3);   // global_prefetch next x panel

        const float* xk = xrow + kp;
#pragma unroll 4
        for (int kk = 0; kk < PANEL_K; kk += 4) {
            v2f a = *(const v2f*)(xk + kk);            // A: 16x4 f32 fragment
#pragma unroll
            for (int n = 0; n < 4; ++n) {
                v2f bv = *(const v2f*)(wp + (n * 16 + r) * WROW + kk + koff);  // B: 4x16
                acc[n] = __builtin_amdgcn_wmma_f32_16x16x4_f32(
                    false, a, false, bv, (short)0, acc[n], false, false);
            }
        }
    }

    // Spill logits tile (C layout: lane->N, vgpr->M row, half selects M+8) to LDS.
    float* mylog = slog + wave * 16 * LROW;
#pragma unroll
    for (int n = 0; n < 4; ++n)
#pragma unroll
        for (int rr = 0; rr < 8; ++rr)
            mylog[(rr + half * 8) * LROW + (n * 16 + r)] = acc[n][rr];

    // Per-token softmax + top-2 (lanes 0..15, one token each).
    if (lane < 16) {
        const int t = lane;
        float* row = mylog + t * LROW;
        const float bias_any = b[0];   // bias is uniform (-log E) but honor buffer anyway
        float mx = -3.4e38f;
        float v1 = -3.4e38f, v2 = -3.4e38f;
        int i1 = 0, i2 = 0;
        for (int e = 0; e < E; ++e) {
            float l = row[e] + b[e];
            row[e] = l;
            mx = fmaxf(mx, l);
            if (l > v1)      { v2 = v1; i2 = i1; v1 = l; i1 = e; }
            else if (l > v2) { v2 = l;  i2 = e; }
        }
        (void)bias_any;
        float sum = 0.0f;
        for (int e = 0; e < E; ++e) {
            float ex = __expf(row[e] - mx);
            row[e] = ex;
            sum += ex;
        }
        float inv = 1.0f / sum;
        row[64] = inv;                 // stash inverse-sum in pad slot
        float p1 = __expf(v1 - mx) * inv;
        float p2 = __expf(v2 - mx) * inv;
        // weights = softmax([p1, p2])
        float a1 = __expf(p1), a2 = __expf(p2);
        float w1 = a1 / (a1 + a2);
        int g = tokBase + t;
        ew[g * 2 + 0] = w1;
        ew[g * 2 + 1] = 1.0f - w1;
        ei[g * 2 + 0] = (float)i1;
        ei[g * 2 + 1] = (float)i2;
        topPacked[g] = ((unsigned)i1 << 8) | (unsigned)i2;
    }

    // Coalesced probs writeback: 16 tokens x 64 experts = contiguous 4KB block.
    float* pdst = probs + (size_t)tokBase * E;
#pragma unroll 4
    for (int i = 0; i < 32; ++i) {
        int flat = i * 32 + lane;
        int t = flat >> 6, e = flat & 63;
        pdst[flat] = mylog[t * LROW + e] * mylog[t * LROW + 64];
    }
}

// ------------- Kernel 2: per-expert capacity mask + aux partials -------------
// One block per expert; deterministic block scans over all T tokens.
__global__ __launch_bounds__(256)
void expert_mask_kernel(const unsigned* __restrict__ topPacked,
                        const float* __restrict__ probs,
                        float* __restrict__ maskOut,    // [E,T]
                        float* __restrict__ partials,   // [E]
                        int T, int cap) {
    const int e = blockIdx.x;
    const int tid = threadIdx.x;
    __shared__ int   sI[256];
    __shared__ float sF[256];

    // Sweep 1: n2_total (top-1 hit count) + prob column sum.
    int cnt = 0; float psum = 0.0f;
    for (int base = 0; base < T; base += 256) {
        unsigned p = topPacked[base + tid];
        cnt  += (int)(((p >> 8) & 0xFFu) == (unsigned)e);
        psum += probs[(size_t)(base + tid) * E + e];
    }
    sI[tid] = cnt; sF[tid] = psum;
    __syncthreads();
    for (int s = 128; s > 0; s >>= 1) {
        if (tid < s) { sI[tid] += sI[tid + s]; sF[tid] += sF[tid + s]; }
        __syncthreads();
    }
    const int n2 = sI[0];
    if (tid == 0) {
        float tf = (float)T;
        partials[e] = (sF[0] / tf) * ((float)n2 / tf) * (float)E * 0.01f;
    }
    __syncthreads();

    // Sweep 2: stable index-ordered ranks -> capacity mask.
    int run2 = 0, run1 = 0;
    for (int base = 0; base < T; base += 256) {
        unsigned p = topPacked[base + tid];
        int f2 = (int)(((p >> 8) & 0xFFu) == (unsigned)e);
        int f1 = (int)((p & 0xFFu) == (unsigned)e);
        int comb = f2 | (f1 << 16);
        __syncthreads();
        sI[tid] = comb;
        __syncthreads();
        for (int off = 1; off < 256; off <<= 1) {     // inclusive Hillis-Steele
            int v = (tid >= off) ? sI[tid - off] : 0;
            __syncthreads();
            sI[tid] += v;
            __syncthreads();
        }
        int incl  = sI[tid];
        int total = sI[255];
        int excl  = incl - comb;
        int rank2 = run2 + (excl & 0xFFFF);
        int rank1 = run1 + (excl >> 16);
        bool sel = f2 ? (rank2 < cap) : (f1 ? (n2 + rank1 < cap) : false);
        maskOut[(size_t)e * T + base + tid] = sel ? 1.0f : 0.0f;
        run2 += total & 0xFFFF;
        run1 += total >> 16;
    }
}

// ---------------- Kernel 3: deterministic aux-loss reduction -----------------
__global__ __launch_bounds__(64)
void aux_reduce_kernel(const float* __restrict__ partials, float* __restrict__ out) {
    __shared__ float s[64];
    int tid = threadIdx.x;
    s[tid] = partials[tid];
    __syncthreads();
    for (int st = 32; st > 0; st >>= 1) {
        if (tid < st) s[tid] += s[tid + st];
        __syncthreads();
    }
    if (tid == 0) *out = s[0];
}

extern "C" void kernel_launch(void* const* d_in, const int* in_sizes, int n_in,
                              void* d_out, int out_size, void* d_ws, size_t ws_size,
                              hipStream_t stream) {
    const float* x = (const float*)d_in[0];   // [B,S,H] f32
    const float* W = (const float*)d_in[1];   // [E,H]   f32
    const float* b = (const float*)d_in[2];   // [E]     f32

    const int h = in_sizes[1] / E;            // 1024
    const int T = in_sizes[0] / h;            // 32768
    (void)n_in; (void)out_size; (void)ws_size;

    // Output layout (flat, return order): ew[T,2] | ei[T,2] | probs[T,64] | aux | mask[E,T]
    float* out  = (float*)d_out;
    float* ew   = out;
    float* ei   = out + (size_t)2 * T;
    float* prb  = out + (size_t)4 * T;
    float* aux  = out + (size_t)(4 + E) * T;
    float* mask = aux + 1;

    unsigned* topPacked = (unsigned*)d_ws;              // T uints
    float* partials = (float*)((char*)d_ws + (size_t)T * sizeof(unsigned));

    int cap = (int)(1.25 * (double)T / (double)E);
    if (cap < 4) cap = 4;

    dim3 g1(T / 64), b1(128);
    router_fused_kernel<<<g1, b1, SMEM_BYTES, stream>>>(x, W, b, ew, ei, prb, topPacked);

    dim3 g2(E), b2(256);
    expert_mask_kernel<<<g2, b2, 0, stream>>>(topPacked, prb, mask, partials, T, cap);

    aux_reduce_kernel<<<1, 64, 0, stream>>>(partials, aux);
}